// Codec_41798621725069
// MI455X (gfx1250) — compile-verified
//
#include <hip/hip_runtime.h>
#include <hip/hip_bf16.h>
#include <stdint.h>

// y[i] = 1.0f - (1.0f - x[i])  (fp32, two rounded subtractions — NOT identity)
//
// Pure streaming kernel: 64 MB traffic @ 23.3 TB/s ~= 2.8 us. Memory-bound at
// 0.25 flop/byte, so no matrix ops; instead use the CDNA5 async global->LDS
// DMA (ASYNCcnt path) for the load side with B128 granularity per lane.

#define BLOCK 256  // 8 wave32s per block

__global__ __launch_bounds__(BLOCK) void Codec_41798621725069_kernel(
    const float* __restrict__ x, float* __restrict__ y, int n) {
  __shared__ float4 smem[BLOCK];

  const int tid = threadIdx.x;
  const long long gid = (long long)blockIdx.x * BLOCK + tid;
  const int n4 = n >> 2;

  if (gid < n4) {
    // Per-lane byte offset into x (fits u32: 8.4M floats * 4B = 33.5 MB < 2^31,
    // so the signed-32-bit VADDR offset of GVS addressing is safe).
    uint32_t goff = (uint32_t)gid * 16u;

    // Raw LDS byte address: generic LDS pointers carry the workgroup-relative
    // LDS offset in their low 32 bits.
    uint32_t lds_addr = (uint32_t)(uintptr_t)(&smem[tid]);

    // CDNA5 async DMA: 16 bytes global -> LDS, tracked by ASYNCcnt.
    // GVS mode: saddr = SGPR64 base, vaddr = per-lane u32 byte offset.
    asm volatile("global_load_async_to_lds_b128 %0, %1, %2"
                 :
                 : "v"(lds_addr), "v"(goff), "s"(x)
                 : "memory");
    asm volatile("s_wait_asynccnt 0" ::: "memory");

    float4 v = smem[tid];  // ds_load_b128
    float4 r;
    r.x = 1.0f - (1.0f - v.x);
    r.y = 1.0f - (1.0f - v.y);
    r.z = 1.0f - (1.0f - v.z);
    r.w = 1.0f - (1.0f - v.w);
    reinterpret_cast<float4*>(y)[gid] = r;  // global_store_b128
  }

  // Scalar tail for n not divisible by 4 (N=8388608 is, but stay general).
  if (blockIdx.x == 0 && tid < (n & 3)) {
    int i = (n4 << 2) + tid;
    y[i] = 1.0f - (1.0f - x[i]);
  }
}

extern "C" void kernel_launch(void* const* d_in, const int* in_sizes, int n_in,
                              void* d_out, int out_size, void* d_ws, size_t ws_size,
                              hipStream_t stream) {
  const float* x = (const float*)d_in[0];
  float* y = (float*)d_out;
  const int n = in_sizes[0];

  const int n4 = n >> 2;
  int blocks = (n4 + BLOCK - 1) / BLOCK;
  if (blocks < 1) blocks = 1;

  Codec_41798621725069_kernel<<<blocks, BLOCK, 0, stream>>>(x, y, n);
}